// Trittention_45260365365946
// MI455X (gfx1250) — compile-verified
//
#include <hip/hip_runtime.h>
#include <hip/hip_bf16.h>

typedef __attribute__((ext_vector_type(2))) float v2f;
typedef __attribute__((ext_vector_type(8))) float v8f;

#define BB 2
#define HH 8
#define SS 160
#define DD 64
#define QB 8              // queries per workgroup (grid = 16 x 20 = 320 WGs)
#define NT 320            // threads per block = 10 waves
#define K1STR 66          // padded LDS row stride (floats) for K tiles
#define ASTR 162          // padded LDS row stride for 160x160 score matrix

// float offsets into dynamic LDS
#define OFF_K1   0
#define OFF_K2   (OFF_K1 + SS*K1STR)       // 10560
#define OFF_A    (OFF_K2 + SS*K1STR)       // 21120
#define OFF_Q    (OFF_A  + SS*ASTR)        // 47040
#define OFF_V1   (OFF_Q  + QB*DD)          // 47552
#define OFF_V2   (OFF_V1 + SS*DD)          // 57792
#define OFF_RS   (OFF_V2 + SS*DD)          // 68032  rowsum partials (NT) / max stage
#define OFF_CS   (OFF_RS + NT)             // 68352  colsum partials (NT)
#define OFF_ROW  (OFF_CS + NT)             // 68672  rowsum (160)
#define OFF_COL  (OFF_ROW + SS)            // 68832  colsum (160)
#define OFF_RED  (OFF_COL + SS)            // 68992  2nd-stage reduce (64)
#define SMEM_FLOATS (OFF_RED + 64)         // 69056 floats = 276224 bytes

__global__ __launch_bounds__(NT) void tritt_kernel(
    const float* __restrict__ gq,  const float* __restrict__ gk1,
    const float* __restrict__ gk2, const float* __restrict__ gv1,
    const float* __restrict__ gv2, float* __restrict__ gz,
    float* __restrict__ glse)
{
    extern __shared__ float sm[];
    float* sK1 = sm + OFF_K1;
    float* sK2 = sm + OFF_K2;
    float* sA  = sm + OFF_A;
    float* sQ  = sm + OFF_Q;
    float* sV1 = sm + OFF_V1;
    float* sV2 = sm + OFF_V2;
    float* sRS = sm + OFF_RS;
    float* sCS = sm + OFF_CS;
    float* sRow= sm + OFF_ROW;
    float* sCol= sm + OFF_COL;
    float* sRed= sm + OFF_RED;

    const int tid  = threadIdx.x;
    const int bh   = blockIdx.x;        // 0..15  (b*H + h)
    const int qb   = blockIdx.y;        // 0..19  query block of QB
    const int lane = tid & 31;
    const int wave = tid >> 5;          // 0..9
    const int base = bh * SS * DD;

    // ---- stage per-(b,h) tensors into LDS ----
    for (int i = tid; i < SS*DD; i += NT) {
        int l = i >> 6, d = i & 63;
        sK1[l*K1STR + d] = gk1[base + i];
        sK2[l*K1STR + d] = gk2[base + i];
        sV1[i] = gv1[base + i];
        sV2[i] = gv2[base + i];
    }
    for (int i = tid; i < QB*DD; i += NT)
        sQ[i] = gq[base + qb*QB*DD + i];
    __syncthreads();

    const int mrow  = lane & 15;            // tile row (A) / tile col (B,C)
    const int khalf = (lane >> 4) << 1;     // 0 or 2: K sub-offset per lane half
    const int chalf = (lane >> 4) << 3;     // 0 or 8: C-matrix M offset per lane half
    const int l0    = wave * 16;            // this wave's l-strip

    for (int qi = 0; qi < QB; ++qi) {
        // ---- Phase A: A[l,r] = (q .* k1[l]) dot k2[r]  via f32 WMMA ----
        v2f a[16];
        #pragma unroll
        for (int kk = 0; kk < 16; ++kk) {
            int kd = (kk << 2) + khalf;
            a[kk].x = sK1[(l0 + mrow)*K1STR + kd]     * sQ[qi*DD + kd];
            a[kk].y = sK1[(l0 + mrow)*K1STR + kd + 1] * sQ[qi*DD + kd + 1];
        }
        for (int rt = 0; rt < 10; ++rt) {
            v8f c = {0.f,0.f,0.f,0.f,0.f,0.f,0.f,0.f};
            #pragma unroll
            for (int kk = 0; kk < 16; ++kk) {
                int kd = (kk << 2) + khalf;
                v2f b;
                b.x = sK2[(rt*16 + mrow)*K1STR + kd];
                b.y = sK2[(rt*16 + mrow)*K1STR + kd + 1];
                c = __builtin_amdgcn_wmma_f32_16x16x4_f32(
                        false, a[kk], false, b, (short)0, c, false, false);
            }
            #pragma unroll
            for (int i = 0; i < 8; ++i)
                sA[(l0 + chalf + i)*ASTR + rt*16 + mrow] = c[i];
        }
        __syncthreads();

        // ---- Phase B: global max over 160x160 ----
        const int row  = tid % SS;
        const int half = tid / SS;          // 0 or 1 -> which 80-col/row slab
        float lmax = -3.4e38f;
        {
            const float* pr = sA + row*ASTR + half*80;
            for (int cI = 0; cI < 80; ++cI) lmax = fmaxf(lmax, pr[cI]);
        }
        sRS[tid] = lmax;
        __syncthreads();
        if (tid < 64) {
            float m = sRS[tid];
            for (int i = tid + 64; i < NT; i += 64) m = fmaxf(m, sRS[i]);
            sRed[tid] = m;
        }
        __syncthreads();
        if (tid == 0) {
            float m = sRed[0];
            for (int i = 1; i < 64; ++i) m = fmaxf(m, sRed[i]);
            sRed[0] = m;
        }
        __syncthreads();
        const float gmax = sRed[0];
        __syncthreads();

        // ---- Phase C1: exp in place + row-sum partials ----
        {
            float s = 0.f;
            float* pr = sA + row*ASTR + half*80;
            for (int cI = 0; cI < 80; ++cI) {
                float e = __expf(pr[cI] - gmax);
                pr[cI] = e;
                s += e;
            }
            sRS[tid] = s;
        }
        __syncthreads();
        // ---- Phase C2: col-sum partials ----
        {
            const int col = tid % SS;
            float s = 0.f;
            for (int rr = 0; rr < 80; ++rr)
                s += sA[(half*80 + rr)*ASTR + col];
            sCS[tid] = s;
        }
        __syncthreads();
        // combine partials; reduce Z from row-sum partials
        if (tid < SS) sRow[tid] = sRS[tid] + sRS[tid + SS];
        else          sCol[tid - SS] = sCS[tid - SS] + sCS[tid];
        if (tid < 64) {
            float z = 0.f;
            for (int i = tid; i < NT; i += 64) z += sRS[i];
            sRed[tid] = z;
        }
        __syncthreads();
        if (tid == 0) {
            float z = 0.f;
            for (int i = 0; i < 64; ++i) z += sRed[i];
            sRed[0] = z;
        }
        __syncthreads();
        const float Z = sRed[0];

        // ---- Phase D: z[q,d] = (rowsum . V1[:,d] + colsum . V2[:,d]) / Z ----
        if (tid < DD) {
            float acc = 0.f;
            for (int l = 0; l < SS; ++l) acc += sRow[l] * sV1[l*DD + tid];
            for (int r = 0; r < SS; ++r) acc += sCol[r] * sV2[r*DD + tid];
            gz[(bh*SS + qb*QB + qi)*DD + tid] = acc / Z;
        }
        if (tid == 0)
            glse[bh*SS + qb*QB + qi] = gmax + __logf(Z);
        __syncthreads();   // protect sA / stats before next query
    }
}

extern "C" void kernel_launch(void* const* d_in, const int* in_sizes, int n_in,
                              void* d_out, int out_size, void* d_ws, size_t ws_size,
                              hipStream_t stream) {
    (void)in_sizes; (void)n_in; (void)out_size; (void)d_ws; (void)ws_size;
    const float* q  = (const float*)d_in[0];
    const float* k1 = (const float*)d_in[1];
    const float* k2 = (const float*)d_in[2];
    const float* v1 = (const float*)d_in[3];
    const float* v2 = (const float*)d_in[4];
    float* z   = (float*)d_out;
    float* lse = z + BB*HH*SS*DD;           // tuple: z then lse, flat

    const size_t smem = SMEM_FLOATS * sizeof(float);  // ~270 KB of 320 KB WGP LDS
    hipFuncSetAttribute(reinterpret_cast<const void*>(tritt_kernel),
                        hipFuncAttributeMaxDynamicSharedMemorySize, (int)smem);
    dim3 grid(BB*HH, SS/QB);                // 16 x 20 = 320 workgroups
    tritt_kernel<<<grid, NT, smem, stream>>>(q, k1, k2, v1, v2, z, lse);
}